// LightGCN_7584912245210
// MI455X (gfx1250) — compile-verified
//
#include <hip/hip_runtime.h>
#include <hip/hip_bf16.h>
#include <stdint.h>

// ---------------- problem constants (match reference) ----------------
enum : int {
    N_USERS = 150000,
    N_ITEMS = 250000,
    N_TOTAL = N_USERS + N_ITEMS,   // 400000
    D       = 64,
    N_HOPS  = 3,
};
enum : long long { NNZ = 8000000LL };

#define EDGES_PER_BLOCK 256
#define OUT_STRIDE      (4 * D)          // [N, 4, 64] layout in d_out
#define MESS_KEEP_SCALE (1.0f / 0.9f)    // 1/(1-MESS_RATE)
#define EDGE_KEEP_SCALE 2.0f             // 1/(1-EDGE_RATE)

typedef float f32x4 __attribute__((ext_vector_type(4)));

// ---------------- CDNA5 async copy helpers (gfx1250) ----------------
// One wave-instruction: each active lane copies 16B global -> 16B LDS,
// tracked by ASYNCcnt (bypasses VGPRs). TH_LOAD_NT: edge stream is
// touched once -> don't evict the L2-resident node table.
__device__ __forceinline__ void async_copy_b128_nt(uint32_t lds_byte_addr, const void* gaddr) {
    asm volatile("global_load_async_to_lds_b128 %0, %1, off th:TH_LOAD_NT"
                 :
                 : "v"(lds_byte_addr), "v"(gaddr)
                 : "memory");
}
__device__ __forceinline__ void wait_asynccnt0() {
    asm volatile("s_wait_asynccnt 0" ::: "memory");
}
// Flat LDS pointers are {aperture_hi, lds_offset}: low 32 bits ARE the LDS
// byte address the async-load VDST VGPR expects (ISA 10.2).
__device__ __forceinline__ uint32_t lds_byte_addr_of(const void* p) {
    return (uint32_t)(uintptr_t)p;
}
// Wave-uniform scalarization helpers (LDS broadcast value -> SGPR).
__device__ __forceinline__ int   rfl_i32(int v)   { return __builtin_amdgcn_readfirstlane(v); }
__device__ __forceinline__ float rfl_f32(float v) {
    return __int_as_float(__builtin_amdgcn_readfirstlane(__float_as_int(v)));
}

// ---------------- kernel 1: init hop-0 slot, zero hop 1..3 slots ----------------
__global__ void lgcn_init_kernel(const float* __restrict__ user_embed,
                                 const float* __restrict__ item_embed,
                                 float* __restrict__ out) {
    int tid = blockIdx.x * blockDim.x + threadIdx.x;     // [0, N_TOTAL*16)
    int n = tid >> 4;
    int q = tid & 15;                                    // float4 slot within row
    if (n >= N_TOTAL) return;

    const float* src = (n < N_USERS)
        ? (user_embed + (size_t)n * D)
        : (item_embed + (size_t)(n - N_USERS) * D);
    // embeddings are read exactly once -> non-temporal
    f32x4 v = __builtin_nontemporal_load((const f32x4*)(src + 4 * q));

    float* dst = out + (size_t)n * OUT_STRIDE + 4 * q;
    *(f32x4*)(dst)         = v;                          // hop 0 (gathered next: keep RT)
    f32x4 z = (f32x4)0.0f;
    *(f32x4*)(dst + 1 * D) = z;                          // hop 1 accumulator
    *(f32x4*)(dst + 2 * D) = z;                          // hop 2 accumulator
    *(f32x4*)(dst + 3 * D) = z;                          // hop 3 accumulator
}

// ---------------- kernel 2: SpMM hop with edge dropout ----------------
// Block = 256 threads = 8 waves = 256 edges. Stage rows/cols/vals/u
// (4 x 1KB) into LDS via async NT b128 copies, then each wave owns one
// edge at a time: lane -> 2 features (float2 gather + 2 f32 atomics).
// Per-edge scalars go through readfirstlane so dropout test / address
// math run on SALU with a scalar branch.
__global__ void lgcn_spmm_kernel(const int*   __restrict__ edge_rows,
                                 const int*   __restrict__ edge_cols,
                                 const float* __restrict__ edge_vals,
                                 const float* __restrict__ edge_u,   // this hop's slice
                                 float* __restrict__ out,
                                 int hop) {
    __shared__ int   s_rows[EDGES_PER_BLOCK];
    __shared__ int   s_cols[EDGES_PER_BLOCK];
    __shared__ float s_vals[EDGES_PER_BLOCK];
    __shared__ float s_u   [EDGES_PER_BLOCK];

    const int tid = threadIdx.x;
    const long long ebase = (long long)blockIdx.x * EDGES_PER_BLOCK;

    // ---- async stage: thread t copies 16B of array (t/64), slot (t%64) ----
    {
        const int a = tid >> 6;
        const int i = tid & 63;
        const long long goff = ebase + 4LL * i;          // 4 elems per 16B copy
        const void* gsrc;
        uint32_t    ldst;
        switch (a) {
            case 0:  gsrc = edge_rows + goff; ldst = lds_byte_addr_of(&s_rows[4 * i]); break;
            case 1:  gsrc = edge_cols + goff; ldst = lds_byte_addr_of(&s_cols[4 * i]); break;
            case 2:  gsrc = edge_vals + goff; ldst = lds_byte_addr_of(&s_vals[4 * i]); break;
            default: gsrc = edge_u    + goff; ldst = lds_byte_addr_of(&s_u   [4 * i]); break;
        }
        async_copy_b128_nt(ldst, gsrc);
        wait_asynccnt0();                // LDS data visible before barrier signal
    }
    __syncthreads();

    // ---- process: wave w handles edges [w*32, w*32+32) of this chunk ----
    const int wv = tid >> 5;
    const int ln = tid & 31;
    const int gather_off  = hop * D + 2 * ln;            // slot h   (read)
    const int scatter_off = (hop + 1) * D + 2 * ln;      // slot h+1 (atomic write)

    for (int j = 0; j < 32; ++j) {
        const int e = wv * 32 + j;
        const float u = rfl_f32(s_u[e]);                 // SGPR -> scalar branch
        if (u >= 0.5f) {                                 // floor(0.5+u)==1 <=> keep
            const int   row = rfl_i32(s_rows[e]);
            const int   col = rfl_i32(s_cols[e]);
            const float v   = rfl_f32(s_vals[e]) * EDGE_KEEP_SCALE;

            // gather: L2-resident (102MB table < 192MB L2, kept hot by NT streams)
            const float2 g = *(const float2*)(out + (size_t)col * OUT_STRIDE + gather_off);

            float* dst = out + (size_t)row * OUT_STRIDE + scatter_off;
            unsafeAtomicAdd(dst,     v * g.x);           // global_atomic_add_f32
            unsafeAtomicAdd(dst + 1, v * g.y);
        }
    }
}

// ---------------- kernel 3: message dropout (inverted) ----------------
__global__ void lgcn_messdrop_kernel(const float* __restrict__ mess_u,  // this hop's slice
                                     float* __restrict__ out,
                                     int hop) {
    int tid = blockIdx.x * blockDim.x + threadIdx.x;     // [0, N_TOTAL*16)
    int n = tid >> 4;
    int q = tid & 15;
    if (n >= N_TOTAL) return;

    // dropout uniforms are read exactly once -> non-temporal
    f32x4 u = __builtin_nontemporal_load((const f32x4*)(mess_u + (size_t)n * D + 4 * q));
    float* dst = out + (size_t)n * OUT_STRIDE + (hop + 1) * D + 4 * q;
    f32x4 a = *(f32x4*)dst;                              // L2 hit (just accumulated)
    a.x = (u.x >= 0.1f) ? a.x * MESS_KEEP_SCALE : 0.0f;
    a.y = (u.y >= 0.1f) ? a.y * MESS_KEEP_SCALE : 0.0f;
    a.z = (u.z >= 0.1f) ? a.z * MESS_KEEP_SCALE : 0.0f;
    a.w = (u.w >= 0.1f) ? a.w * MESS_KEEP_SCALE : 0.0f;
    *(f32x4*)dst = a;                                    // re-gathered next hop: keep RT
}

// ---------------- host launch ----------------
extern "C" void kernel_launch(void* const* d_in, const int* in_sizes, int n_in,
                              void* d_out, int out_size, void* d_ws, size_t ws_size,
                              hipStream_t stream) {
    const float* user_embed  = (const float*)d_in[0];
    const float* item_embed  = (const float*)d_in[1];
    const int*   edge_rows   = (const int*)  d_in[2];
    const int*   edge_cols   = (const int*)  d_in[3];
    const float* edge_vals   = (const float*)d_in[4];
    const float* edge_drop_u = (const float*)d_in[5];   // [3, NNZ]
    const float* mess_drop_u = (const float*)d_in[6];   // [3, N_TOTAL, D]
    float* out = (float*)d_out;                         // [N_TOTAL, 4, 64] flat

    const int vec_blocks  = (N_TOTAL * 16) / 256;          // 25000, exact
    const int spmm_blocks = (int)(NNZ / EDGES_PER_BLOCK);  // 31250, exact

    lgcn_init_kernel<<<vec_blocks, 256, 0, stream>>>(user_embed, item_embed, out);

    for (int hop = 0; hop < N_HOPS; ++hop) {
        lgcn_spmm_kernel<<<spmm_blocks, 256, 0, stream>>>(
            edge_rows, edge_cols, edge_vals,
            edge_drop_u + (size_t)hop * NNZ,
            out, hop);
        lgcn_messdrop_kernel<<<vec_blocks, 256, 0, stream>>>(
            mess_drop_u + (size_t)hop * N_TOTAL * D,
            out, hop);
    }
}